// GAT_49289044689249
// MI455X (gfx1250) — compile-verified
//
#include <hip/hip_runtime.h>
#include <math.h>

typedef float v2f __attribute__((ext_vector_type(2)));
typedef float v8f __attribute__((ext_vector_type(8)));

#define GAT_N      50000
#define GAT_E      800000
#define GAT_ET     (GAT_E + GAT_N)
#define GAT_INC    128
#define GAT_HID    32
#define GAT_HEADS  8
#define GAT_OUTC   32
#define NEG_SLOPE  0.2f

// ---------------------------------------------------------------------------
// WMMA fp32 GEMM: C[nrows x ncol] = A[nrows x K] * B[K x ncol]
// One wave32 computes one 16x16 tile; inner loop issues V_WMMA_F32_16X16X4_F32.
// A-layout per lane: lanes 0-15 -> (M=l15, K=k+0/1), lanes 16-31 -> (M=l15, K=k+2/3)
// B-layout per lane: lanes 0-15 -> (K=k+0/1, N=l15), lanes 16-31 -> (K=k+2/3, N=l15)
// D-layout: acc[r] -> (M = r + half*8, N = l15)
// ---------------------------------------------------------------------------
__global__ void gat_wmma_gemm(const float* __restrict__ A,
                              const float* __restrict__ B,
                              float* __restrict__ C,
                              int K, int ncol, int tilesN, int totalTiles) {
    int wave = (int)((blockIdx.x * blockDim.x + threadIdx.x) >> 5);
    if (wave >= totalTiles) return;           // uniform per wave: EXEC stays all-1s
    int lane = threadIdx.x & 31;
    int half = lane >> 4;
    int l15  = lane & 15;
    int tm = wave / tilesN;
    int tn = wave % tilesN;

    const float* arow = A + (long)(tm * 16 + l15) * K + 2 * half;
    const float* bcol = B + (long)(2 * half) * ncol + tn * 16 + l15;

    v8f acc = {0.f, 0.f, 0.f, 0.f, 0.f, 0.f, 0.f, 0.f};
    for (int k = 0; k < K; k += 4) {
        v2f a = *reinterpret_cast<const v2f*>(arow + k);
        v2f b;
        b.x = bcol[(long)k * ncol];
        b.y = bcol[(long)k * ncol + ncol];
        acc = __builtin_amdgcn_wmma_f32_16x16x4_f32(
            /*neg_a=*/false, a, /*neg_b=*/false, b,
            /*c_mod=*/(short)0, acc, /*reuse_a=*/false, /*reuse_b=*/false);
    }

    float* crow = C + (long)(tm * 16 + half * 8) * ncol + tn * 16 + l15;
#pragma unroll
    for (int r = 0; r < 8; ++r)
        crow[(long)r * ncol] = acc[r];
}

// ---------------------------------------------------------------------------
// Per-(node, head) attention scores: s_src / s_dst dot products over channels
// ---------------------------------------------------------------------------
__global__ void gat_scores(const float* __restrict__ H,
                           const float* __restrict__ att_s,
                           const float* __restrict__ att_d,
                           float* __restrict__ Ss, float* __restrict__ Sd,
                           int n, int heads, int ch) {
    int i = blockIdx.x * blockDim.x + threadIdx.x;
    if (i >= n * heads) return;
    int h = i % heads;
    const float* hv = H + (long)i * ch;
    float ss = 0.f, sd = 0.f;
    for (int c = 0; c < ch; ++c) {
        float v = hv[c];
        ss += v * att_s[h * ch + c];
        sd += v * att_d[h * ch + c];
    }
    Ss[i] = ss;
    Sd[i] = sd;
}

__global__ void gat_fill(float* __restrict__ p, float v, long n) {
    long i = blockIdx.x * (long)blockDim.x + threadIdx.x;
    if (i < n) p[i] = v;
}

__device__ __forceinline__ void atomicMaxFloat(float* addr, float v) {
    // sign-split trick: valid for IEEE fp32 ordering, init must be -inf
    if (v >= 0.f)
        atomicMax((int*)addr, __float_as_int(v));
    else
        atomicMin((unsigned int*)addr, (unsigned int)__float_as_int(v));
}

__device__ __forceinline__ float gat_leaky(float v) {
    return v > 0.f ? v : NEG_SLOPE * v;
}

// edge id e: e < E -> (ei[e], ei[E+e]); e >= E -> self loop (e-E, e-E)
__device__ __forceinline__ void gat_edge(const long long* __restrict__ ei,
                                         long e, long& s, long& d) {
    if (e < GAT_E) {
        s = (long)ei[e];
        d = (long)ei[GAT_E + e];
    } else {
        s = e - GAT_E;
        d = s;
    }
}

// ---------------------------------------------------------------------------
// Pass 1: segment max of leaky_relu(s_src[src]+s_dst[dst]) grouped by dst
// ---------------------------------------------------------------------------
__global__ void gat_edge_max(const long long* __restrict__ ei,
                             const float* __restrict__ Ss,
                             const float* __restrict__ Sd,
                             float* __restrict__ dmax, int heads) {
    long idx = blockIdx.x * (long)blockDim.x + threadIdx.x;
    if (idx >= (long)GAT_ET * heads) return;
    int h = (int)(idx % heads);
    long e = idx / heads;
    long s, d;
    gat_edge(ei, e, s, d);
    float v = gat_leaky(Ss[s * heads + h] + Sd[d * heads + h]);
    atomicMaxFloat(&dmax[d * heads + h], v);
}

// ---------------------------------------------------------------------------
// Pass 2: denominator sum of exp(e - max[dst])
// ---------------------------------------------------------------------------
__global__ void gat_edge_expsum(const long long* __restrict__ ei,
                                const float* __restrict__ Ss,
                                const float* __restrict__ Sd,
                                const float* __restrict__ dmax,
                                float* __restrict__ den, int heads) {
    long idx = blockIdx.x * (long)blockDim.x + threadIdx.x;
    if (idx >= (long)GAT_ET * heads) return;
    int h = (int)(idx % heads);
    long e = idx / heads;
    long s, d;
    gat_edge(ei, e, s, d);
    float v = gat_leaky(Ss[s * heads + h] + Sd[d * heads + h]);
    float ex = expf(v - dmax[d * heads + h]);
    atomicAdd(&den[d * heads + h], ex);
}

// ---------------------------------------------------------------------------
// Pass 3: agg[dst,h,c] += alpha * H[src,h,c]   (one lane per channel)
// ---------------------------------------------------------------------------
__global__ void gat_edge_agg(const long long* __restrict__ ei,
                             const float* __restrict__ Ss,
                             const float* __restrict__ Sd,
                             const float* __restrict__ dmax,
                             const float* __restrict__ den,
                             const float* __restrict__ H,
                             float* __restrict__ agg, int heads, int ch) {
    long idx = blockIdx.x * (long)blockDim.x + threadIdx.x;
    if (idx >= (long)GAT_ET * heads * ch) return;
    int c = (int)(idx % ch);
    long eh = idx / ch;
    int h = (int)(eh % heads);
    long e = eh / heads;
    long s, d;
    gat_edge(ei, e, s, d);
    long dh = d * heads + h;
    float v = gat_leaky(Ss[s * heads + h] + Sd[dh]);
    float alpha = expf(v - dmax[dh]) / den[dh];
    atomicAdd(&agg[dh * ch + c], alpha * H[(s * heads + h) * (long)ch + c]);
}

// out = elu(agg + bias)   (in place)
__global__ void gat_bias_elu(float* __restrict__ A, const float* __restrict__ b,
                             long n, int cols) {
    long i = blockIdx.x * (long)blockDim.x + threadIdx.x;
    if (i >= n) return;
    float v = A[i] + b[i % cols];
    A[i] = v > 0.f ? v : (expf(v) - 1.0f);
}

// ---------------------------------------------------------------------------
// Final: out[n,:] = log_softmax(agg2[n,:] + b2) over 32 classes (one wave/node)
// ---------------------------------------------------------------------------
__global__ void gat_logsoftmax(const float* __restrict__ agg,
                               const float* __restrict__ b,
                               float* __restrict__ out, int n) {
    int wave = (int)((blockIdx.x * blockDim.x + threadIdx.x) >> 5);
    int lane = threadIdx.x & 31;
    if (wave >= n) return;
    float v = agg[(long)wave * 32 + lane] + b[lane];
    float m = v;
#pragma unroll
    for (int o = 16; o > 0; o >>= 1) m = fmaxf(m, __shfl_xor(m, o, 32));
    float ex = expf(v - m);
    float sum = ex;
#pragma unroll
    for (int o = 16; o > 0; o >>= 1) sum += __shfl_xor(sum, o, 32);
    out[(long)wave * 32 + lane] = v - m - logf(sum);
}

// ---------------------------------------------------------------------------
extern "C" void kernel_launch(void* const* d_in, const int* in_sizes, int n_in,
                              void* d_out, int out_size, void* d_ws, size_t ws_size,
                              hipStream_t stream) {
    const float*     x        = (const float*)d_in[0];
    const long long* ei       = (const long long*)d_in[1];   // int64 [2,E]
    const float*     W1       = (const float*)d_in[2];
    const float*     att_src1 = (const float*)d_in[3];
    const float*     att_dst1 = (const float*)d_in[4];
    const float*     b1       = (const float*)d_in[5];
    const float*     W2       = (const float*)d_in[6];
    const float*     att_src2 = (const float*)d_in[7];
    const float*     att_dst2 = (const float*)d_in[8];
    const float*     b2       = (const float*)d_in[9];
    float* out = (float*)d_out;

    const int F1 = GAT_HEADS * GAT_HID;   // 256

    // workspace layout (floats)
    float* ws   = (float*)d_ws;
    float* H1   = ws;                                   // N*256
    float* AGG1 = H1   + (long)GAT_N * F1;              // N*256 (becomes ELU feat)
    float* S1s  = AGG1 + (long)GAT_N * F1;              // N*8
    float* S1d  = S1s  + (long)GAT_N * GAT_HEADS;
    float* DM1  = S1d  + (long)GAT_N * GAT_HEADS;
    float* DEN1 = DM1  + (long)GAT_N * GAT_HEADS;
    float* H2   = DEN1 + (long)GAT_N * GAT_HEADS;       // N*32
    float* AGG2 = H2   + (long)GAT_N * GAT_OUTC;        // N*32
    float* S2s  = AGG2 + (long)GAT_N * GAT_OUTC;        // N
    float* S2d  = S2s  + GAT_N;
    float* DM2  = S2d  + GAT_N;
    float* DEN2 = DM2  + GAT_N;

    const int TB = 256;
    auto blocks = [](long work, int tb) { return (unsigned)((work + tb - 1) / tb); };

    // zero accumulators (graph-capture-safe async memset)
    hipMemsetAsync(DEN1, 0, (long)GAT_N * GAT_HEADS * sizeof(float), stream);
    hipMemsetAsync(AGG1, 0, (long)GAT_N * F1 * sizeof(float), stream);
    hipMemsetAsync(DEN2, 0, (long)GAT_N * sizeof(float), stream);
    hipMemsetAsync(AGG2, 0, (long)GAT_N * GAT_OUTC * sizeof(float), stream);
    gat_fill<<<blocks((long)GAT_N * GAT_HEADS, TB), TB, 0, stream>>>(
        DM1, -INFINITY, (long)GAT_N * GAT_HEADS);
    gat_fill<<<blocks(GAT_N, TB), TB, 0, stream>>>(DM2, -INFINITY, GAT_N);

    // ---------------- layer 1 ----------------
    {
        int tilesN = F1 / 16;                      // 16
        int totalTiles = (GAT_N / 16) * tilesN;    // 50000
        gat_wmma_gemm<<<blocks((long)totalTiles * 32, TB), TB, 0, stream>>>(
            x, W1, H1, GAT_INC, F1, tilesN, totalTiles);

        gat_scores<<<blocks((long)GAT_N * GAT_HEADS, TB), TB, 0, stream>>>(
            H1, att_src1, att_dst1, S1s, S1d, GAT_N, GAT_HEADS, GAT_HID);

        long eh = (long)GAT_ET * GAT_HEADS;
        gat_edge_max<<<blocks(eh, TB), TB, 0, stream>>>(ei, S1s, S1d, DM1, GAT_HEADS);
        gat_edge_expsum<<<blocks(eh, TB), TB, 0, stream>>>(ei, S1s, S1d, DM1, DEN1, GAT_HEADS);
        gat_edge_agg<<<blocks(eh * GAT_HID, TB), TB, 0, stream>>>(
            ei, S1s, S1d, DM1, DEN1, H1, AGG1, GAT_HEADS, GAT_HID);

        gat_bias_elu<<<blocks((long)GAT_N * F1, TB), TB, 0, stream>>>(
            AGG1, b1, (long)GAT_N * F1, F1);
    }

    // ---------------- layer 2 ----------------
    {
        int tilesN = GAT_OUTC / 16;                // 2
        int totalTiles = (GAT_N / 16) * tilesN;    // 6250
        gat_wmma_gemm<<<blocks((long)totalTiles * 32, TB), TB, 0, stream>>>(
            AGG1, W2, H2, F1, GAT_OUTC, tilesN, totalTiles);

        gat_scores<<<blocks(GAT_N, TB), TB, 0, stream>>>(
            H2, att_src2, att_dst2, S2s, S2d, GAT_N, 1, GAT_OUTC);

        long eh = (long)GAT_ET;
        gat_edge_max<<<blocks(eh, TB), TB, 0, stream>>>(ei, S2s, S2d, DM2, 1);
        gat_edge_expsum<<<blocks(eh, TB), TB, 0, stream>>>(ei, S2s, S2d, DM2, DEN2, 1);
        gat_edge_agg<<<blocks(eh * GAT_OUTC, TB), TB, 0, stream>>>(
            ei, S2s, S2d, DM2, DEN2, H2, AGG2, 1, GAT_OUTC);

        gat_logsoftmax<<<blocks((long)GAT_N * 32, TB), TB, 0, stream>>>(
            AGG2, b2, out, GAT_N);
    }
}